// OutputLayer_36344013259152
// MI455X (gfx1250) — compile-verified
//
#include <hip/hip_runtime.h>

// MI455X / gfx1250. Bandwidth-bound conv1x1-reduce (378 MB streamed once).
// Reduction done with V_WMMA_F32_16X16X4_F32 (full fp32 -> matches fp32 ref).

typedef float v2f __attribute__((ext_vector_type(2)));
typedef float v8f __attribute__((ext_vector_type(8)));

__device__ __forceinline__ int bias_index(int hw) {
    // D4-orbit index for a 19x19 board, matching build_bias_map().
    int i = hw / 19;
    int j = hw - i * 19;
    int a = i < 18 - i ? i : 18 - i;
    int b = j < 18 - j ? j : 18 - j;
    int ci = a < b ? a : b;
    int cj = a < b ? b : a;
    int rank = ci * 10 - (ci * (ci - 1)) / 2 + (cj - ci);
    return (rank + 54) % 55;   // (counter - 1) mod 55, Python semantics
}

__global__ __launch_bounds__(256)
void conv1x1_bias_wmma(const float* __restrict__ x,      // [1024,256,361]
                       const float* __restrict__ w,      // [256]
                       const float* __restrict__ bias,   // [55]
                       float* __restrict__ out) {        // [1024*361]
    __shared__ float wlds[256];
    wlds[threadIdx.x] = w[threadIdx.x];
    __syncthreads();

    const int lane  = threadIdx.x & 31;
    const int wave  = threadIdx.x >> 5;
    const int obase = (blockIdx.x * 8 + wave) * 16;   // 16 outputs per wave

    // A-matrix layout (16x4 f32): V0 = K0 (lanes 0-15) / K2 (lanes 16-31),
    //                             V1 = K1 / K3.
    const int m  = lane & 15;            // output row M
    const int kk = (lane >> 4) << 1;     // 0 for lanes 0-15, 2 for lanes 16-31
    const int o  = obase + m;            // flat output index = b*361 + hw
    const int b  = o / 361;
    const int hw = o - b * 361;
    const float* pA = x + (size_t)b * 92416 + (size_t)kk * 361 + hw;

    v8f acc = {};
    #pragma unroll 4
    for (int k = 0; k < 256; k += 4) {
        v2f a;
        a.x = __builtin_nontemporal_load(pA);        // x[b, k+kk,   hw]
        a.y = __builtin_nontemporal_load(pA + 361);  // x[b, k+kk+1, hw]
        // B-matrix (4x16 broadcast of w): lanes 0-15 need w[k],w[k+1];
        // lanes 16-31 need w[k+2],w[k+3]. k+kk is even -> 8B-aligned b64 load.
        v2f bv = *(const v2f*)(wlds + k + kk);
        acc = __builtin_amdgcn_wmma_f32_16x16x4_f32(
            /*neg_a=*/false, a, /*neg_b=*/false, bv,
            /*c_mod=*/(short)0, acc, /*reuse_a=*/false, /*reuse_b=*/false);
        pA += 4 * 361;
    }

    // All N columns of D are identical. D VGPR r = row M=r (lanes 0-15) or
    // M=r+8 (lanes 16-31): lane 0 stores rows 0-7, lane 16 stores rows 8-15.
    if (lane == 0 || lane == 16) {
        const int rowoff = (lane >> 4) * 8;
        #pragma unroll
        for (int r = 0; r < 8; ++r) {
            int oo = obase + rowoff + r;
            int bb = oo / 361;
            int hh = oo - bb * 361;
            out[oo] = acc[r] + bias[bias_index(hh)];
        }
    }
}

extern "C" void kernel_launch(void* const* d_in, const int* in_sizes, int n_in,
                              void* d_out, int out_size, void* d_ws, size_t ws_size,
                              hipStream_t stream) {
    const float* x    = (const float*)d_in[0];   // (1024,256,19,19) fp32
    const float* w    = (const float*)d_in[1];   // (1,256,1,1) fp32
    const float* bias = (const float*)d_in[2];   // (55,) fp32
    float* out = (float*)d_out;                  // (1024,361) fp32

    // 1024*361 = 369,664 outputs; 128 outputs per 256-thread block -> exact fit.
    int blocks = out_size / 128;                 // 2888
    conv1x1_bias_wmma<<<blocks, 256, 0, stream>>>(x, w, bias, out);
}